// ImprovedNNConv_5385888989817
// MI455X (gfx1250) — compile-verified
//
#include <hip/hip_runtime.h>
#include <math.h>

typedef float v2f __attribute__((ext_vector_type(2)));
typedef float v8f __attribute__((ext_vector_type(8)));

#define HDIM 64
#define EDIM 16
#define INDIM 32
#define OUTDIM 6
#define HH2 (HDIM * HDIM)  /* 4096 */
#define JSTR 20            /* LDS floats per We column: 16 data + 4 pad; 4096*20*4B = 320KB */
#define BN_EPS 1e-5f

// D = A(16x4) * B(4x16) + C, all f32 (CDNA5 WMMA, wave32)
__device__ __forceinline__ v8f wmma4(v2f a, v2f b, v8f c) {
  return __builtin_amdgcn_wmma_f32_16x16x4_f32(false, a, false, b, (short)0, c,
                                               false, false);
}
__device__ __forceinline__ float relu1(float x) {
  return __builtin_amdgcn_fmed3f(x, 0.0f, 3.0e38f);  // single v_med3_f32
}

// ---------------- node encoder: h = x @ Wx + bx ----------------
__global__ void enc_kernel(const float* __restrict__ x, const float* __restrict__ Wx,
                           const float* __restrict__ bx, float* __restrict__ h, int N) {
  int t = blockIdx.x * blockDim.x + threadIdx.x;
  if (t >= N * HDIM) return;
  int n = t >> 6, j = t & 63;
  const float* xr = x + (size_t)n * INDIM;
  float s = bx[j];
#pragma unroll
  for (int k = 0; k < INDIM; ++k) s += xr[k] * Wx[k * HDIM + j];
  h[t] = s;
}

// ---------------- destination degree, then 1/max(deg,1) ----------------
__global__ void deg_kernel(const int* __restrict__ dst, float* __restrict__ cnt, int E) {
  int t = blockIdx.x * blockDim.x + threadIdx.x;
  if (t < E) atomicAdd(&cnt[dst[t]], 1.0f);
}
__global__ void inv_kernel(float* __restrict__ cnt, int N) {
  int t = blockIdx.x * blockDim.x + threadIdx.x;
  if (t < N) cnt[t] = 1.0f / fmaxf(cnt[t], 1.0f);
}

// ---------------- fused edge kernel ----------------
// Per 16-edge block: t^T = We^T(4096x16) @ ea^T(16x16) via WMMA (rows = MLP
// outputs j, cols = edges). We^T staged in LDS K-permuted so each lane's
// A-fragment is 8 contiguous floats (2x ds_load_b128/tile). Bias folded into
// C as one v8f broadcast load; relu via v_med3; scale by h[src][hh] (float4
// per 4 tiles); msg^T accumulated in 8 VGPRs; atomic scatter into agg[dst].
__global__ __launch_bounds__(512) void edge_kernel(
    const float* __restrict__ ea, const int* __restrict__ src,
    const int* __restrict__ dst, const float* __restrict__ h,
    const float* __restrict__ We_l, const float* __restrict__ be_l,
    float* __restrict__ agg, int E) {
  extern __shared__ float lds[];  // HH2 * JSTR floats (exactly 320 KB)

  // permuted cooperative stage: lds[j*JSTR + h*8 + kk*2 + v] = We_l[(kk*4+2h+v)*4096 + j]
  for (int idx = threadIdx.x * 4; idx < EDIM * HH2; idx += blockDim.x * 4) {
    int k = idx >> 12;            // We row (0..15)
    int j = idx & (HH2 - 1);      // We column (multiple of 4)
    int pos = ((k >> 1) & 1) * 8 + (k >> 2) * 2 + (k & 1);
    float4 w = *(const float4*)(We_l + idx);
    lds[(j + 0) * JSTR + pos] = w.x;
    lds[(j + 1) * JSTR + pos] = w.y;
    lds[(j + 2) * JSTR + pos] = w.z;
    lds[(j + 3) * JSTR + pos] = w.w;
  }
  __syncthreads();

  const int lane = threadIdx.x & 31;
  const int wave = threadIdx.x >> 5;
  const int ln   = lane & 15;   // tile column = edge-in-block
  const int half = lane >> 4;   // 0/1
  const int team = wave >> 2;   // 4 edge-block teams per WG (16 waves)
  const int wt   = wave & 3;    // o-tile (16 of 64 output features)
  const int nblk = (E + 15) >> 4;

  for (int blk = blockIdx.x * 4 + team; blk < nblk; blk += gridDim.x * 4) {
    int e = blk * 16 + ln;
    bool valid = e < E;
    int ec = valid ? e : (E - 1);

    // B fragments: edge_attr^T (K=16 in 4 chunks of 4)
    v2f bf[4];
#pragma unroll
    for (int kk = 0; kk < 4; ++kk)
      bf[kk] = *(const v2f*)(ea + (size_t)ec * EDIM + kk * 4 + 2 * half);

    const float* hrow = h + (size_t)src[ec] * HDIM;
    int dn = dst[ec];

    v8f acc = {};  // msg^T[wt*16 + r + 8*half][edge ln]
    for (int hh4 = 0; hh4 < HDIM; hh4 += 4) {
      float4 hs4 = *(const float4*)(hrow + hh4);
      const float hsv[4] = {hs4.x, hs4.y, hs4.z, hs4.w};
#pragma unroll
      for (int u = 0; u < 4; ++u) {
        const int hh = hh4 + u;
        const int j0 = hh * HDIM + wt * 16;
        const float* ap = lds + (size_t)(j0 + ln) * JSTR + half * 8;
        float4 a01 = *(const float4*)(ap);      // chunks 0,1
        float4 a23 = *(const float4*)(ap + 4);  // chunks 2,3
        v8f d = *(const v8f*)(be_l + j0 + 8 * half);  // bias as C-init (broadcast)
        v2f a;
        a[0] = a01.x; a[1] = a01.y; d = wmma4(a, bf[0], d);
        a[0] = a01.z; a[1] = a01.w; d = wmma4(a, bf[1], d);
        a[0] = a23.x; a[1] = a23.y; d = wmma4(a, bf[2], d);
        a[0] = a23.z; a[1] = a23.w; d = wmma4(a, bf[3], d);
        const float hs = hsv[u];
#pragma unroll
        for (int r = 0; r < 8; ++r) acc[r] += hs * relu1(d[r]);
      }
    }

    if (valid) {
      float* ag = agg + (size_t)dn * HDIM + wt * 16 + 8 * half;
#pragma unroll
      for (int r = 0; r < 8; ++r) atomicAdd(&ag[r], acc[r]);
    }
  }
}

// ---------------- node kernel: out = relu(agg/deg + h@Wroot + broot); BN; +h ----
__global__ __launch_bounds__(256) void node_kernel(
    const float* __restrict__ h, const float* __restrict__ agg,
    const float* __restrict__ invc, const float* __restrict__ Wr_l,
    const float* __restrict__ br_l, const float* __restrict__ gam,
    const float* __restrict__ bet, const float* __restrict__ mu,
    const float* __restrict__ var, float* __restrict__ hout, int N) {
  const int lane = threadIdx.x & 31;
  const int wave = threadIdx.x >> 5;
  const int ln   = lane & 15;
  const int half = lane >> 4;
  const int team = wave >> 2;
  const int wt   = wave & 3;
  const int o    = wt * 16 + ln;

  // Wroot fragments (K=64 -> 16 chunks); fixed per wave, hoisted out of the loop
  v2f bf[16];
#pragma unroll
  for (int kk = 0; kk < 16; ++kk) {
    int k0 = kk * 4 + 2 * half;
    bf[kk][0] = Wr_l[(k0)     * HDIM + o];
    bf[kk][1] = Wr_l[(k0 + 1) * HDIM + o];
  }
  float sc  = gam[o] * rsqrtf(var[o] + BN_EPS);
  float sh  = bet[o] - mu[o] * sc;
  float bro = br_l[o];

  const int nblk = (N + 15) >> 4;
  for (int blk = blockIdx.x * 2 + team; blk < nblk; blk += gridDim.x * 2) {
    int n0 = blk * 16;
    int nrow = n0 + ln;
    if (nrow >= N) nrow = N - 1;  // clamp A loads, no EXEC divergence around WMMA
    v8f d;
#pragma unroll
    for (int r = 0; r < 8; ++r) d[r] = bro;
#pragma unroll
    for (int kk = 0; kk < 16; ++kk) {
      v2f a = *(const v2f*)(h + (size_t)nrow * HDIM + kk * 4 + 2 * half);
      d = wmma4(a, bf[kk], d);
    }
#pragma unroll
    for (int r = 0; r < 8; ++r) {
      int row = n0 + r + 8 * half;
      if (row < N) {
        size_t idx = (size_t)row * HDIM + o;
        float m = d[r] + agg[idx] * invc[row];
        m = relu1(m);
        m = m * sc + sh;
        hout[idx] = m + h[idx];
      }
    }
  }
}

// ---------------- global mean pool (sums + counts) ----------------
__global__ void pool_kernel(const float* __restrict__ h, const int* __restrict__ batch,
                            float* __restrict__ pool, float* __restrict__ cntb, int N) {
  int t = blockIdx.x * blockDim.x + threadIdx.x;
  if (t >= N * HDIM) return;
  int n = t >> 6, j = t & 63;
  int g = batch[n];
  atomicAdd(&pool[g * HDIM + j], h[t]);
  if (j == 0) atomicAdd(&cntb[g], 1.0f);
}

// ---------------- head: z = relu(pooled@W1+b1); out = z@W2+b2 ----------------
__global__ __launch_bounds__(256) void head_kernel(
    const float* __restrict__ pool, const float* __restrict__ cntb,
    const float* __restrict__ W1, const float* __restrict__ b1,
    const float* __restrict__ W2, const float* __restrict__ b2,
    float* __restrict__ out, int G) {
  __shared__ float pl[64 * HDIM];
  __shared__ float zl[64 * HDIM];
  int GG = G > 64 ? 64 : G;
  for (int t = threadIdx.x; t < GG * HDIM; t += blockDim.x) {
    int g = t >> 6;
    pl[t] = pool[t] / fmaxf(cntb[g], 1.0f);
  }
  __syncthreads();
  for (int t = threadIdx.x; t < GG * HDIM; t += blockDim.x) {
    int g = t >> 6, j = t & 63;
    float s = b1[j];
#pragma unroll 8
    for (int k = 0; k < HDIM; ++k) s += pl[g * HDIM + k] * W1[k * HDIM + j];
    zl[t] = relu1(s);
  }
  __syncthreads();
  for (int t = threadIdx.x; t < GG * OUTDIM; t += blockDim.x) {
    int g = t / OUTDIM, od = t % OUTDIM;
    float s = b2[od];
#pragma unroll 8
    for (int k = 0; k < HDIM; ++k) s += zl[g * HDIM + k] * W2[k * OUTDIM + od];
    out[t] = s;
  }
}

extern "C" void kernel_launch(void* const* d_in, const int* in_sizes, int n_in,
                              void* d_out, int out_size, void* d_ws, size_t ws_size,
                              hipStream_t stream) {
  (void)n_in; (void)ws_size;
  const float* x   = (const float*)d_in[0];
  const float* ea  = (const float*)d_in[1];
  const int*   src = (const int*)d_in[2];
  const int*   dst = (const int*)d_in[3];
  const int*   bat = (const int*)d_in[4];
  const float* Wx  = (const float*)d_in[5];
  const float* bx  = (const float*)d_in[6];
  const float* We  = (const float*)d_in[7];   // (3, 16, 4096)
  const float* be  = (const float*)d_in[8];   // (3, 4096)
  const float* Wr  = (const float*)d_in[9];   // (3, 64, 64)
  const float* br  = (const float*)d_in[10];  // (3, 64)
  const float* gam = (const float*)d_in[11];
  const float* bet = (const float*)d_in[12];
  const float* mu  = (const float*)d_in[13];
  const float* var = (const float*)d_in[14];
  const float* W1  = (const float*)d_in[15];
  const float* b1  = (const float*)d_in[16];
  const float* W2  = (const float*)d_in[17];
  const float* b2  = (const float*)d_in[18];
  float* out = (float*)d_out;

  const int N = in_sizes[0] / INDIM;
  const int E = in_sizes[1] / EDIM;
  const int G = out_size / OUTDIM;

  float* ws   = (float*)d_ws;
  size_t NH   = (size_t)N * HDIM;
  float* h0   = ws;
  float* h1   = h0 + NH;
  float* agg  = h1 + NH;
  float* invc = agg + NH;
  float* pool = invc + N;
  float* cntb = pool + (size_t)G * HDIM;

  hipMemsetAsync(invc, 0, (size_t)N * sizeof(float), stream);
  hipMemsetAsync(pool, 0, (size_t)G * HDIM * sizeof(float), stream);
  hipMemsetAsync(cntb, 0, (size_t)G * sizeof(float), stream);

  const size_t EDGE_LDS = (size_t)HH2 * JSTR * sizeof(float);  // 327,680 B = 320 KB
  hipFuncSetAttribute((const void*)edge_kernel,
                      hipFuncAttributeMaxDynamicSharedMemorySize, (int)EDGE_LDS);

  const int thr = 256;
  enc_kernel<<<(N * HDIM + thr - 1) / thr, thr, 0, stream>>>(x, Wx, bx, h0, N);
  deg_kernel<<<(E + thr - 1) / thr, thr, 0, stream>>>(dst, invc, E);
  inv_kernel<<<(N + thr - 1) / thr, thr, 0, stream>>>(invc, N);

  const float* hcur = h0;
  float* hnxt = h1;
  int nblkE = (E + 15) / 16;
  int egrid = (nblkE + 3) / 4; if (egrid > 512)  egrid = 512;
  int nblkN = (N + 15) / 16;
  int ngrid = (nblkN + 1) / 2; if (ngrid > 1024) ngrid = 1024;

  for (int l = 0; l < 3; ++l) {
    hipMemsetAsync(agg, 0, NH * sizeof(float), stream);
    edge_kernel<<<egrid, 512, EDGE_LDS, stream>>>(
        ea, src, dst, hcur, We + (size_t)l * EDIM * HH2, be + (size_t)l * HH2, agg, E);
    node_kernel<<<ngrid, 256, 0, stream>>>(
        hcur, agg, invc, Wr + (size_t)l * HDIM * HDIM, br + (size_t)l * HDIM,
        gam + l * HDIM, bet + l * HDIM, mu + l * HDIM, var + l * HDIM, hnxt, N);
    float* t = (float*)hcur; hcur = hnxt; hnxt = t;
  }

  pool_kernel<<<(N * HDIM + thr - 1) / thr, thr, 0, stream>>>(hcur, bat, pool, cntb, N);
  head_kernel<<<1, 256, 0, stream>>>(pool, cntb, W1, b1, W2, b2, out, G);
}